// FreqFilter_61031485276217
// MI455X (gfx1250) — compile-verified
//
#include <hip/hip_runtime.h>

// FreqFilter: out = Dh^T @ ((Dh @ X @ Dw^T) * A_c) @ Dw  per (b,c) 5x5 tile.
// Recast per channel c as GEMM: Out_c[4096x25] = X_c[4096x25] @ M_c^T[25x25],
// where M_c is precomputed from freq_att (kernel 1, stored in d_ws in WMMA
// B-fragment layout, zero-padded K->28 / N->32). Kernel 2 streams x exactly
// once and uses V_WMMA_F32_16X16X4_F32 (fp32 WMMA; kernel is HBM-bound at
// ~524 MB total traffic, so fp32 matches the reference at zero cost).

typedef __attribute__((ext_vector_type(2))) float v2f;
typedef __attribute__((ext_vector_type(8))) float v8f;

namespace {
constexpr int kC     = 640;
constexpr int kHW    = 25;   // 5x5 tile, flattened
constexpr int kSteps = 7;    // K padded 25 -> 28, 4 per WMMA step
// per-channel B storage: 7 K-steps x 2 N-tiles x 2 VGPRs x 32 lanes
constexpr int kBPerChan = kSteps * 2 * 2 * 32;  // 896 floats
constexpr int kRowStride = kC * kHW;            // 16000 floats between batches
}

// Orthonormal 5-point DCT-II matrix D[k][i] (row 0 scaled by 1/sqrt2).
__device__ __constant__ float kDCT[5][5] = {
  { 0.44721359550f,  0.44721359550f,  0.44721359550f,  0.44721359550f,  0.44721359550f },
  { 0.60150095500f,  0.37174803446f,  0.00000000000f, -0.37174803446f, -0.60150095500f },
  { 0.51166727361f, -0.19543950758f, -0.63245553203f, -0.19543950758f,  0.51166727361f },
  { 0.37174803446f, -0.60150095500f,  0.00000000000f,  0.60150095500f, -0.37174803446f },
  { 0.19543950758f, -0.51166727361f,  0.63245553203f, -0.51166727361f,  0.19543950758f },
};

// Kernel 1: one wave per channel. Builds M_c^T directly in the WMMA B-matrix
// register layout:  B[k][n] with k = 4*s + v + 2*h (v = vgpr, h = lane>>4),
// n = lane&15; N-tile nt selects columns nt*16+n. Zero for k>=25 or o>=25
// (this zero-padding is what lets kernel 2 feed don't-care A values at K>=25).
__global__ void build_b_kernel(const float* __restrict__ att,
                               float* __restrict__ bmat) {
  const int c    = blockIdx.x;
  const int lane = threadIdx.x & 31;
  const int n    = lane & 15;
  const int h    = lane >> 4;
  const float* A = att + c * kHW;          // freq_att[c][5][5]
  float* out     = bmat + (size_t)c * kBPerChan;

  for (int nt = 0; nt < 2; ++nt) {
    const int o  = nt * 16 + n;            // output flat index (B column)
    const int oh = o / 5, ow = o % 5;
    for (int s = 0; s < kSteps; ++s) {
      for (int v = 0; v < 2; ++v) {
        const int k = 4 * s + v + 2 * h;   // input flat index (B row)
        float val = 0.0f;
        if (k < kHW && o < kHW) {
          const int ih = k / 5, iw = k % 5;
          #pragma unroll
          for (int p = 0; p < 5; ++p) {
            float tp = 0.0f;
            #pragma unroll
            for (int q = 0; q < 5; ++q)
              tp += kDCT[q][ow] * A[p * 5 + q] * kDCT[q][iw];
            val += kDCT[p][oh] * kDCT[p][ih] * tp;
          }
        }
        out[((s * 2 + nt) * 2 + v) * 32 + lane] = val;
      }
    }
  }
}

// Kernel 2: grid = (80 channel-groups, 32 chunk-slots), block = 256 (8 waves).
// Wave w handles channel blockIdx.x*8+w; all 8 waves of a block walk the SAME
// 16-batch chunks, so the block's global reads cover contiguous 800-byte spans
// (8 adjacent 100-byte tiles) per batch row -> full DRAM sector utilization,
// boundary lines shared via WGP$/L2. Per chunk: 14 x v_wmma_f32_16x16x4_f32.
__global__ void __launch_bounds__(256)
freq_filter_kernel(const float* __restrict__ x,
                   const float* __restrict__ bmat,
                   float* __restrict__ y) {
  const int wave = threadIdx.x >> 5;
  const int lane = threadIdx.x & 31;
  const int n    = lane & 15;
  const int h    = lane >> 4;
  const int c    = blockIdx.x * 8 + wave;

  // B fragments for this channel (2.3 MB total -> resident in L2),
  // amortized over 8 chunks = 112 WMMAs.
  const float* bp = bmat + (size_t)c * kBPerChan;
  v2f bf0[kSteps], bf1[kSteps];
  #pragma unroll
  for (int s = 0; s < kSteps; ++s) {
    bf0[s].x = bp[((s * 2 + 0) * 2 + 0) * 32 + lane];
    bf0[s].y = bp[((s * 2 + 0) * 2 + 1) * 32 + lane];
    bf1[s].x = bp[((s * 2 + 1) * 2 + 0) * 32 + lane];
    bf1[s].y = bp[((s * 2 + 1) * 2 + 1) * 32 + lane];
  }

  for (int it = 0; it < 8; ++it) {
    const int chunk = blockIdx.y + 32 * it;        // 0..255 (16-batch chunk)
    // A-matrix lane row: M = lane&15 (both halves), per ISA 16x4 f32 layout.
    const float* xr = x + ((size_t)(chunk * 16 + n) * kC + c) * kHW;

    v8f acc0 = {};
    v8f acc1 = {};
    // Steps 0..5: lane's K pair k0 = 4s+2h, k0+1 (all < 24, unconditional).
    #pragma unroll
    for (int s = 0; s < 6; ++s) {
      const int k0 = 4 * s + 2 * h;
      v2f a;
      a.x = xr[k0];
      a.y = xr[k0 + 1];
      acc0 = __builtin_amdgcn_wmma_f32_16x16x4_f32(
                 false, a, false, bf0[s], (short)0, acc0, false, false);
      acc1 = __builtin_amdgcn_wmma_f32_16x16x4_f32(
                 false, a, false, bf1[s], (short)0, acc1, false, false);
    }
    // Step 6: K = 24..27. Only K=24 (h=0, v=0) is live; B rows 25..27 are
    // zero, so h=1 / a.y values are don't-cares -> one unconditional load,
    // in-bounds address for every lane, no exec-mask branch.
    {
      v2f a;
      a.x = xr[h ? 0 : 24];
      a.y = 0.0f;
      acc0 = __builtin_amdgcn_wmma_f32_16x16x4_f32(
                 false, a, false, bf0[6], (short)0, acc0, false, false);
      acc1 = __builtin_amdgcn_wmma_f32_16x16x4_f32(
                 false, a, false, bf1[6], (short)0, acc1, false, false);
    }

    // C/D layout: VGPR r -> (M=r, N=lane) lanes 0-15; (M=r+8, N=lane-16).
    // One per-lane base pointer; r*16000 floats = r*64000 B folds into the
    // signed 24-bit instruction offset. Non-temporal: y is write-once.
    float* yp = y + ((size_t)(chunk * 16 + 8 * h) * kC + c) * kHW + n;
    #pragma unroll
    for (int r = 0; r < 8; ++r)
      __builtin_nontemporal_store(acc0[r], yp + (size_t)r * kRowStride);
    if (n < 9) {                                   // single exec region
      #pragma unroll
      for (int r = 0; r < 8; ++r)
        __builtin_nontemporal_store(acc1[r], yp + (size_t)r * kRowStride + 16);
    }
  }
}

extern "C" void kernel_launch(void* const* d_in, const int* in_sizes, int n_in,
                              void* d_out, int out_size, void* d_ws, size_t ws_size,
                              hipStream_t stream) {
  (void)in_sizes; (void)n_in; (void)out_size; (void)ws_size;
  const float* x   = (const float*)d_in[0];
  const float* att = (const float*)d_in[1];
  float* y         = (float*)d_out;
  float* bmat      = (float*)d_ws;   // needs 640*896*4 = 2,293,760 bytes

  build_b_kernel<<<dim3(kC), dim3(32), 0, stream>>>(att, bmat);
  freq_filter_kernel<<<dim3(kC / 8, 32), dim3(256), 0, stream>>>(x, bmat, y);
}